// Unpatch_87299505258572
// MI455X (gfx1250) — compile-verified
//
#include <hip/hip_runtime.h>

// Unpatch: patches[B=32, gj=8, gi=8, ph=64, pw=64, C=3] f32 (contiguous)
//       -> images [B=32, H=512, W=512, C=3] f32
//
// image[b, j*64+p, i*64+q, c] = patches[b, j, i, p, q, c]
//
// Pure permutation, 0 FLOPs, ~201 MB HBM traffic -> bandwidth bound
// (~8.6 us floor at 23.3 TB/s on MI455X). For fixed (b,j,i,p) the (q,c)
// run of 192 floats (768 B) is contiguous on BOTH sides: the op is
// 131072 contiguous 768 B row-segment copies.
//
// Mapping: one 192-thread block (6 wave32) per (b,j,i) patch = 64 segments.
//  - thread t: sl = t/48 (sub-segment), w = t%48 (float4 within segment)
//  - input float4 index  = blockIdx*3072 + t          (block reads are
//    perfectly sequential: sl*48 + w == t)
//  - output float4 index = ((blockIdx>>3)*64 + p)*384 + (blockIdx&7)*48 + w,
//    p = sl + 4*round
// Per-round strides are compile-time constants (+3072 B in, +24576 B out),
// so after full unroll the body is 32 b128 vmem ops with immediate offsets
// and zero per-iteration address math. NT cache hints: in+out = 201 MB >
// 192 MB L2, so stream both directions instead of thrashing.

typedef float __attribute__((ext_vector_type(4))) v4f;

__global__ __launch_bounds__(192) void unpatch_copy(
    const float* __restrict__ in, float* __restrict__ out)
{
    const unsigned t  = threadIdx.x;       // 0..191
    const unsigned b  = blockIdx.x;        // 0..2047  == (b*8 + j)*8 + i
    const unsigned sl = t / 48u;           // 0..3   (one division, at entry)
    const unsigned w  = t - sl * 48u;      // 0..47

    const unsigned i  = b & 7u;            // grid-x index
    const unsigned bj = b >> 3;            // b*8 + j

    // float4-granular base offsets
    const unsigned in4  = b * 3072u + t;                      // == (b*64+sl)*48 + w
    const unsigned out4 = (bj * 64u + sl) * 384u + i * 48u + w;

    const v4f* __restrict__ pin  = (const v4f*)in  + in4;
    v4f*       __restrict__ pout = (v4f*)out + out4;

    // 16 rounds (p = sl, sl+4, ..., sl+60): 4 outer iterations, each with
    // 4 independent NT b128 loads then 4 NT b128 stores. Full unroll turns
    // every offset into an instruction immediate.
#pragma unroll
    for (int r = 0; r < 4; ++r) {
        const v4f* p = pin  + r * 768;     // 4 rounds * 192 float4  (12288 B)
        v4f*       q = pout + r * 6144;    // 4 rounds * 1536 float4 (98304 B)

        v4f d0 = __builtin_nontemporal_load(p);
        v4f d1 = __builtin_nontemporal_load(p + 192);
        v4f d2 = __builtin_nontemporal_load(p + 384);
        v4f d3 = __builtin_nontemporal_load(p + 576);

        __builtin_nontemporal_store(d0, q);
        __builtin_nontemporal_store(d1, q + 1536);
        __builtin_nontemporal_store(d2, q + 3072);
        __builtin_nontemporal_store(d3, q + 4608);
    }
}

extern "C" void kernel_launch(void* const* d_in, const int* in_sizes, int n_in,
                              void* d_out, int out_size, void* d_ws, size_t ws_size,
                              hipStream_t stream) {
    const float* in  = (const float*)d_in[0];
    float*       out = (float*)d_out;

    // 2048 blocks x 192 threads x 16 float4 = 25,165,824 floats (exact cover)
    dim3 block(192);
    dim3 grid(2048);
    unpatch_copy<<<grid, block, 0, stream>>>(in, out);
}